// GATLayer_84722524881413
// MI455X (gfx1250) — compile-verified
//
#include <hip/hip_runtime.h>

typedef float v2f __attribute__((ext_vector_type(2)));
typedef float v4f __attribute__((ext_vector_type(4)));
typedef float v8f __attribute__((ext_vector_type(8)));
typedef int   v4i __attribute__((ext_vector_type(4)));

constexpr int B = 8, N = 1024, D = 256, R = 12;
constexpr float ALPHA   = 0.2f;
constexpr float NEG_INF = -9.0e15f;

#define DEVFN static __device__ __forceinline__

DEVFN float lrelu(float x) { return x >= 0.f ? x : ALPHA * x; }
DEVFN float eluf (float x) { return x > 0.f ? x : __expf(x) - 1.f; }

DEVFN v8f wmma_f32(v2f a, v2f b, v8f c) {
  // D = A(16x4,f32) * B(4x16,f32) + C(16x16,f32)
  return __builtin_amdgcn_wmma_f32_16x16x4_f32(
      false, a, false, b, (short)0, c, false, false);
}

// ---------------------------------------------------------------------------
// K1: Wh = H @ W for both h0 and h1.  One wave => 16 rows x 64 cols.
// A frag: lane<16 holds K={k,k+1}, lane>=16 holds K={k+2,k+3} for row (lane&15).
// ---------------------------------------------------------------------------
__global__ __launch_bounds__(256) void k_wh(
    const float* __restrict__ h0, const float* __restrict__ h1,
    const float* __restrict__ W,
    float* __restrict__ wh0, float* __restrict__ wh1)
{
  const int wave  = blockIdx.x * 8 + (threadIdx.x >> 5);  // 4096 waves
  const int lane  = threadIdx.x & 31;
  const int nblk  = wave & 3;                  // 64-col strip
  const int mtile = wave >> 2;                 // 0..1023 over 16384 rows
  const int rowg  = mtile * 16;
  const bool sel  = rowg >= B * N;
  const int row   = rowg & (B * N - 1);
  const float* __restrict__ H = sel ? h1 : h0;
  float* __restrict__ O       = sel ? wh1 : wh0;

  const int m    = lane & 15;
  const int koff = (lane < 16) ? 0 : 2;
  const float* __restrict__ Arow = H + (size_t)(row + m) * D;
  const int nb = nblk * 64 + m;

  v8f acc0 = {0,0,0,0,0,0,0,0}, acc1 = {0,0,0,0,0,0,0,0};
  v8f acc2 = {0,0,0,0,0,0,0,0}, acc3 = {0,0,0,0,0,0,0,0};

  for (int k = 0; k < D; k += 4) {
    v2f a; a.x = Arow[k + koff]; a.y = Arow[k + koff + 1];
    const float* Wp = W + (size_t)(k + koff) * D + nb;
    v2f b0; b0.x = Wp[0];  b0.y = Wp[D];
    v2f b1; b1.x = Wp[16]; b1.y = Wp[D + 16];
    v2f b2; b2.x = Wp[32]; b2.y = Wp[D + 32];
    v2f b3; b3.x = Wp[48]; b3.y = Wp[D + 48];
    acc0 = wmma_f32(a, b0, acc0);
    acc1 = wmma_f32(a, b1, acc1);
    acc2 = wmma_f32(a, b2, acc2);
    acc3 = wmma_f32(a, b3, acc3);
  }

  const int rbase = row + ((lane < 16) ? 0 : 8);
  #pragma unroll
  for (int v = 0; v < 8; ++v) {
    size_t o = (size_t)(rbase + v) * D + nblk * 64 + m;
    O[o]      = acc0[v];
    O[o + 16] = acc1[v];
    O[o + 32] = acc2[v];
    O[o + 48] = acc3[v];
  }
}

// ---------------------------------------------------------------------------
// K2: per-row projections s{0,1}[row] and r{0,1}[r][row].  One wave per row.
// ---------------------------------------------------------------------------
__global__ __launch_bounds__(256) void k_proj(
    const float* __restrict__ wh0, const float* __restrict__ wh1,
    const float* __restrict__ avec, const float* __restrict__ A_rel,
    float* __restrict__ s0, float* __restrict__ s1,
    float* __restrict__ r0, float* __restrict__ r1)
{
  const int wave = blockIdx.x * 8 + (threadIdx.x >> 5);   // 16384 waves
  const int lane = threadIdx.x & 31;
  const int sel  = (wave >= B * N) ? 1 : 0;
  const int row  = wave & (B * N - 1);
  const float* __restrict__ Wh = sel ? wh1 : wh0;
  const float* __restrict__ av = avec + sel * D;
  const float* __restrict__ Ar = A_rel + sel * D;   // row stride 2*D

  float acc[R + 1];
  #pragma unroll
  for (int t = 0; t <= R; ++t) acc[t] = 0.f;

  const float* __restrict__ x = Wh + (size_t)row * D;
  for (int it = 0; it < D / 32; ++it) {
    const int d = lane + it * 32;
    const float v = x[d];
    acc[0] = fmaf(v, av[d], acc[0]);
    #pragma unroll
    for (int r = 0; r < R; ++r)
      acc[1 + r] = fmaf(v, Ar[r * 2 * D + d], acc[1 + r]);
  }
  #pragma unroll
  for (int t = 0; t <= R; ++t) {
    float v = acc[t];
    v += __shfl_xor(v, 16, 32);
    v += __shfl_xor(v,  8, 32);
    v += __shfl_xor(v,  4, 32);
    v += __shfl_xor(v,  2, 32);
    v += __shfl_xor(v,  1, 32);
    acc[t] = v;
  }
  if (lane == 0) {
    float* sdst = sel ? s1 : s0;
    float* rdst = sel ? r1 : r0;
    sdst[row] = acc[0];
    #pragma unroll
    for (int r = 0; r < R; ++r) rdst[r * (B * N) + row] = acc[1 + r];
  }
}

// ---------------------------------------------------------------------------
// K3: logits[b,i,j] — the bandwidth-dominant pass (streams 402MB multi_adj).
// Block = one (b,i) row; thread = 4 consecutive j (b128 NT loads).
// ---------------------------------------------------------------------------
__global__ __launch_bounds__(256) void k_logits(
    const int* __restrict__ multi_adj, const int* __restrict__ adj,
    const float* __restrict__ s0, const float* __restrict__ s1,
    const float* __restrict__ r0, const float* __restrict__ r1,
    float* __restrict__ logits)
{
  const int bi = blockIdx.x;          // b*N + i
  const int b  = bi >> 10;
  const int j0 = threadIdx.x * 4;

  const float s0i = s0[bi];
  const v4f s1v = *reinterpret_cast<const v4f*>(s1 + b * N + j0);
  const v4i adj4 = __builtin_nontemporal_load(
      reinterpret_cast<const v4i*>(adj + (size_t)bi * N + j0));

  v4f att;
  #pragma unroll
  for (int c = 0; c < 4; ++c) {
    const float e = lrelu(s0i + s1v[c]);
    att[c] = (adj4[c] > 0) ? e : NEG_INF;
  }

  v4f rel = {0.f, 0.f, 0.f, 0.f};
  #pragma unroll
  for (int r = 0; r < R; ++r) {
    const float r0v = r0[r * (B * N) + bi];
    const v4f r1v = *reinterpret_cast<const v4f*>(r1 + r * (B * N) + b * N + j0);
    const v4i m4 = __builtin_nontemporal_load(
        reinterpret_cast<const v4i*>(multi_adj + ((size_t)r * B * N + bi) * N + j0));
    #pragma unroll
    for (int c = 0; c < 4; ++c)
      if (m4[c] > 0) rel[c] += lrelu(r0v + r1v[c]);
  }

  v4f outv;
  #pragma unroll
  for (int c = 0; c < 4; ++c) outv[c] = att[c] + 0.01f * rel[c];
  *reinterpret_cast<v4f*>(logits + (size_t)bi * N + j0) = outv;
}

// ---------------------------------------------------------------------------
// K4a/K4b: column softmax stats over axis i (per (b,j)), chunked for occupancy.
// ---------------------------------------------------------------------------
__global__ __launch_bounds__(256) void k_stats_part(
    const float* __restrict__ logits, float* __restrict__ pm, float* __restrict__ ps)
{
  const int id = blockIdx.x;          // 8 b * 4 jblk * 4 chunk = 128
  const int c  = id & 3;
  const int jb = (id >> 2) & 3;
  const int b  = id >> 4;
  const int j  = jb * 256 + threadIdx.x;
  const float* __restrict__ Lg = logits + (size_t)b * N * N + j;

  float mM = -3.0e38f, sS = 0.f;
  for (int i = c * 256; i < c * 256 + 256; ++i) {
    const float x  = Lg[(size_t)i * N];
    const float nm = fmaxf(mM, x);
    sS = sS * __expf(mM - nm) + __expf(x - nm);
    mM = nm;
  }
  pm[c * (B * N) + b * N + j] = mM;
  ps[c * (B * N) + b * N + j] = sS;
}

__global__ __launch_bounds__(256) void k_stats_comb(
    const float* __restrict__ pm, const float* __restrict__ ps,
    float* __restrict__ cm, float* __restrict__ cs)
{
  const int g = blockIdx.x * 256 + threadIdx.x;   // 8192 columns
  float M = -3.0e38f;
  #pragma unroll
  for (int c = 0; c < 4; ++c) M = fmaxf(M, pm[c * (B * N) + g]);
  float S = 0.f;
  #pragma unroll
  for (int c = 0; c < 4; ++c) S += ps[c * (B * N) + g] * __expf(pm[c * (B * N) + g] - M);
  cm[g] = M;
  cs[g] = 1.f / S;
}

// ---------------------------------------------------------------------------
// K5: out = elu( softmax_col(logits) @ Wh1 ).  att computed on the fly into
// the WMMA A-fragment.  One wave => 16 rows x 64 cols, K=1024 in steps of 4.
// ---------------------------------------------------------------------------
__global__ __launch_bounds__(256) void k_attmm(
    const float* __restrict__ logits, const float* __restrict__ cm,
    const float* __restrict__ cs, const float* __restrict__ wh1,
    float* __restrict__ out)
{
  const int wave  = blockIdx.x * 8 + (threadIdx.x >> 5);  // 2048 waves
  const int lane  = threadIdx.x & 31;
  const int nblk  = wave & 3;
  const int mtile = (wave >> 2) & 63;
  const int b     = wave >> 8;

  const int m    = lane & 15;
  const int koff = (lane < 16) ? 0 : 2;
  const int row0 = mtile * 16;

  const float* __restrict__ Lrow = logits + ((size_t)b * N + row0 + m) * N;
  const float* __restrict__ M    = cm + b * N;
  const float* __restrict__ S    = cs + b * N;
  const float* __restrict__ Wh   = wh1 + (size_t)b * N * D;
  const int nb = nblk * 64 + m;

  v8f acc0 = {0,0,0,0,0,0,0,0}, acc1 = {0,0,0,0,0,0,0,0};
  v8f acc2 = {0,0,0,0,0,0,0,0}, acc3 = {0,0,0,0,0,0,0,0};

  for (int k = 0; k < N; k += 4) {
    const int c0 = k + koff, c1 = c0 + 1;
    v2f a;
    a.x = __expf(Lrow[c0] - M[c0]) * S[c0];
    a.y = __expf(Lrow[c1] - M[c1]) * S[c1];
    const float* Wp = Wh + (size_t)c0 * D + nb;
    v2f b0; b0.x = Wp[0];  b0.y = Wp[D];
    v2f b1; b1.x = Wp[16]; b1.y = Wp[D + 16];
    v2f b2; b2.x = Wp[32]; b2.y = Wp[D + 32];
    v2f b3; b3.x = Wp[48]; b3.y = Wp[D + 48];
    acc0 = wmma_f32(a, b0, acc0);
    acc1 = wmma_f32(a, b1, acc1);
    acc2 = wmma_f32(a, b2, acc2);
    acc3 = wmma_f32(a, b3, acc3);
  }

  const int rbase = row0 + ((lane < 16) ? 0 : 8);
  #pragma unroll
  for (int v = 0; v < 8; ++v) {
    size_t o = ((size_t)b * N + rbase + v) * D + nblk * 64 + m;
    out[o]      = eluf(acc0[v]);
    out[o + 16] = eluf(acc1[v]);
    out[o + 32] = eluf(acc2[v]);
    out[o + 48] = eluf(acc3[v]);
  }
}

// ---------------------------------------------------------------------------
// Host launch
// ---------------------------------------------------------------------------
extern "C" void kernel_launch(void* const* d_in, const int* in_sizes, int n_in,
                              void* d_out, int out_size, void* d_ws, size_t ws_size,
                              hipStream_t stream) {
  (void)in_sizes; (void)n_in; (void)out_size; (void)ws_size;

  const float* h0        = (const float*)d_in[0];
  const float* h1        = (const float*)d_in[1];
  const int*   multi_adj = (const int*)  d_in[2];
  const int*   adj       = (const int*)  d_in[3];
  const float* W         = (const float*)d_in[4];
  const float* avec      = (const float*)d_in[5];
  const float* A_rel     = (const float*)d_in[6];
  float* out = (float*)d_out;
  char*  ws  = (char*)d_ws;

  constexpr size_t WH_BYTES  = (size_t)B * N * D * 4;     // 8 MiB
  constexpr size_t S_BYTES   = (size_t)B * N * 4;         // 32 KiB
  constexpr size_t R_BYTES   = (size_t)R * B * N * 4;     // 384 KiB
  constexpr size_t P_BYTES   = (size_t)4 * B * N * 4;     // 128 KiB
  constexpr size_t OFF_WH0 = 0;
  constexpr size_t OFF_WH1 = OFF_WH0 + WH_BYTES;
  constexpr size_t OFF_S0  = OFF_WH1 + WH_BYTES;
  constexpr size_t OFF_S1  = OFF_S0 + S_BYTES;
  constexpr size_t OFF_R0  = OFF_S1 + S_BYTES;
  constexpr size_t OFF_R1  = OFF_R0 + R_BYTES;
  constexpr size_t OFF_PM  = OFF_R1 + R_BYTES;
  constexpr size_t OFF_PS  = OFF_PM + P_BYTES;
  constexpr size_t OFF_CM  = OFF_PS + P_BYTES;
  constexpr size_t OFF_CS  = OFF_CM + S_BYTES;
  constexpr size_t OFF_LOG = (OFF_CS + S_BYTES + 255) & ~(size_t)255;

  float* wh0 = (float*)(ws + OFF_WH0);
  float* wh1 = (float*)(ws + OFF_WH1);
  float* s0  = (float*)(ws + OFF_S0);
  float* s1  = (float*)(ws + OFF_S1);
  float* r0  = (float*)(ws + OFF_R0);
  float* r1  = (float*)(ws + OFF_R1);
  float* pm  = (float*)(ws + OFF_PM);
  float* ps  = (float*)(ws + OFF_PS);
  float* cmv = (float*)(ws + OFF_CM);
  float* csv = (float*)(ws + OFF_CS);
  float* lg  = (float*)(ws + OFF_LOG);

  k_wh        <<<512,  256, 0, stream>>>(h0, h1, W, wh0, wh1);
  k_proj      <<<2048, 256, 0, stream>>>(wh0, wh1, avec, A_rel, s0, s1, r0, r1);
  k_logits    <<<B * N, 256, 0, stream>>>(multi_adj, adj, s0, s1, r0, r1, lg);
  k_stats_part<<<128,  256, 0, stream>>>(lg, pm, ps);
  k_stats_comb<<<32,   256, 0, stream>>>(pm, ps, cmv, csv);
  k_attmm     <<<256,  256, 0, stream>>>(lg, cmv, csv, wh1, out);
}